// PrismRegularizationLoss_75917841924417
// MI455X (gfx1250) — compile-verified
//
#include <hip/hip_runtime.h>
#include <hip/hip_bf16.h>

typedef float v2f __attribute__((ext_vector_type(2)));
typedef float v8f __attribute__((ext_vector_type(8)));

#define BLOCK 256
#define WAVES 8   // 256 / wave32

// coeff[p][q] = 2^-(|p0-q0|+|p1-q1|) with p=(p>>1, p&1)
__device__ __forceinline__ float coeff_val(int p, int q) {
    int d = abs((p >> 1) - (q >> 1)) + abs((p & 1) - (q & 1));
    return 1.0f / (float)(1 << d);
}

// M12 = coeff (x) I3, zero-padded to 16x16.  Branchless -> v_cndmask.
__device__ __forceinline__ float m12(int k, int n) {
    int same = (k < 12) & (n < 12) & ((k % 3) == (n % 3));
    float v = coeff_val(k / 3, n / 3);
    return same ? v : 0.0f;
}

__device__ __forceinline__ void cross3(const float* a, const float* b, float* o) {
    o[0] = a[1] * b[2] - a[2] * b[1];
    o[1] = a[2] * b[0] - a[0] * b[2];
    o[2] = a[0] * b[1] - a[1] * b[0];
}

__device__ __forceinline__ float face_area(const float* __restrict__ v,
                                           int i0, int i1, int i2) {
    float u[3], w[3], c[3];
    #pragma unroll
    for (int k = 0; k < 3; ++k) {
        u[k] = v[3 * i1 + k] - v[3 * i0 + k];
        w[k] = v[3 * i2 + k] - v[3 * i0 + k];
    }
    cross3(u, w, c);
    return 0.5f * sqrtf(c[0] * c[0] + c[1] * c[1] + c[2] * c[2]);
}

__global__ void zero_out_kernel(float* out) {
    if (threadIdx.x == 0 && blockIdx.x == 0) out[0] = 0.0f;
}

__global__ void __launch_bounds__(BLOCK)
prism_loss_kernel(const float* __restrict__ tp,     // (F,3,3)
                  const float* __restrict__ rot,    // (F,3,3)
                  const float* __restrict__ verts,  // (V,3)
                  const int*   __restrict__ faces,  // (3,F)  Ft[f][j] = faces[j*F+f]
                  const int*   __restrict__ edges,  // (E,2)
                  const int*   __restrict__ fids,   // (E,2)
                  float* __restrict__ out, int E, int F)
{
    __shared__ float Dbuf[WAVES][32][13];      // 16x12 X tiles (row = edge, col = k)
    __shared__ float Ybuf[WAVES][2][16][17];   // Y = X*M12 per 16-edge group

    const int tid  = threadIdx.x;
    const int wave = tid >> 5;
    const int lane = tid & 31;
    const int m    = lane & 15;   // M (pass1 A) / N (pass1 B) index
    const int half = lane >> 4;

    const int gidx  = blockIdx.x * BLOCK + tid;
    const bool valid = (gidx < E);
    const int e = valid ? gidx : (E - 1);   // clamp: keep EXEC all-1s for WMMA

    const int e0 = edges[2 * e + 0];
    const int e1 = edges[2 * e + 1];
    const int f1 = fids[2 * e + 0];
    const int f2 = fids[2 * e + 1];

    // ---- gather prisms + rotations for both faces ----
    float t1[9], t2[9], r1[9], r2[9];
    #pragma unroll
    for (int i = 0; i < 9; ++i) {
        t1[i] = tp[f1 * 9 + i];  t2[i] = tp[f2 * 9 + i];
        r1[i] = rot[f1 * 9 + i]; r2[i] = rot[f2 * 9 + i];
    }

    // ---- unit normals from transformed prisms ----
    float n1[3], n2[3];
    {
        float a[3], b[3], c[3];
        #pragma unroll
        for (int k = 0; k < 3; ++k) { a[k] = t1[k] - t1[3 + k]; b[k] = t1[k] - t1[6 + k]; }
        cross3(a, b, c);
        float inv = rsqrtf(c[0] * c[0] + c[1] * c[1] + c[2] * c[2]);
        #pragma unroll
        for (int k = 0; k < 3; ++k) n1[k] = c[k] * inv;
        #pragma unroll
        for (int k = 0; k < 3; ++k) { a[k] = t2[k] - t2[3 + k]; b[k] = t2[k] - t2[6 + k]; }
        cross3(a, b, c);
        inv = rsqrtf(c[0] * c[0] + c[1] * c[1] + c[2] * c[2]);
        #pragma unroll
        for (int k = 0; k < 3; ++k) n2[k] = c[k] * inv;
    }

    // ---- locate edge endpoints inside each face (JAX -1 wraps to idx 2) ----
    const int fa0 = faces[0 * F + f1], fa1 = faces[1 * F + f1], fa2 = faces[2 * F + f1];
    const int fb0 = faces[0 * F + f2], fb1 = faces[1 * F + f2], fb2 = faces[2 * F + f2];
    const int a0 = (fa0 == e0) ? 0 : ((fa1 == e0) ? 1 : 2);
    const int a1 = (fa0 == e1) ? 0 : ((fa1 == e1) ? 1 : 2);
    const int b0 = (fb0 == e0) ? 0 : ((fb1 == e0) ? 1 : 2);
    const int b1 = (fb0 == e1) ? 0 : ((fb1 == e1) ? 1 : 2);

    // ---- rotated normals: out[d] = sum_c n[c]*R[c][d] ----
    float p1n1[3], p1n2[3], p2n1[3], p2n2[3];
    #pragma unroll
    for (int d = 0; d < 3; ++d) {
        p1n1[d] = n1[0] * r1[d] + n1[1] * r1[3 + d] + n1[2] * r1[6 + d];
        p1n2[d] = n2[0] * r1[d] + n2[1] * r1[3 + d] + n2[2] * r1[6 + d];
        p2n1[d] = n1[0] * r2[d] + n1[1] * r2[3 + d] + n1[2] * r2[6 + d];
        p2n2[d] = n2[0] * r2[d] + n2[1] * r2[3 + d] + n2[2] * r2[6 + d];
    }

    // ---- D = A - B (4x3), flattened d12[p*3+c], H = 1 ----
    float d12[12];
    #pragma unroll
    for (int c = 0; c < 3; ++c) {
        float u1 = t1[a0 * 3 + c] - t2[b0 * 3 + c];
        float u2 = t1[a1 * 3 + c] - t2[b1 * 3 + c];
        float w1 = p1n1[c] - p2n1[c];
        float w2 = p1n2[c] - p2n2[c];
        d12[0 + c] = u1 + w1;
        d12[3 + c] = u2 + w2;
        d12[6 + c] = u1 - w1;
        d12[9 + c] = u2 - w2;
    }

    // ---- weight = |v[e0]-v[e1]|^2 / (area(f1)+area(f2)) ----
    float wv = 0.0f;
    #pragma unroll
    for (int c = 0; c < 3; ++c) {
        float dv = verts[3 * e0 + c] - verts[3 * e1 + c];
        wv += dv * dv;
    }
    const float areaSum = face_area(verts, fa0, fa1, fa2) + face_area(verts, fb0, fb1, fb2);
    const float weight = valid ? (wv / areaSum) : 0.0f;

    // ---- stage X tiles in LDS ----
    #pragma unroll
    for (int k = 0; k < 12; ++k) Dbuf[wave][lane][k] = d12[k];
    __syncthreads();

    // ---- Y = X * M12 via V_WMMA_F32_16X16X4_F32, K=12 in three 4-slices ----
    // A layout (f32 16x4): lane = M (mod 16), half selects K {k0,k0+1} vs {k0+2,k0+3}
    // B layout (f32 4x16): lane = N (mod 16), half selects the same K split
    #pragma unroll
    for (int g = 0; g < 2; ++g) {
        v8f acc = {0.f, 0.f, 0.f, 0.f, 0.f, 0.f, 0.f, 0.f};
        #pragma unroll
        for (int s = 0; s < 3; ++s) {
            const int k0 = 4 * s + 2 * half;
            v2f av, bv;
            av.x = Dbuf[wave][g * 16 + m][k0 + 0];
            av.y = Dbuf[wave][g * 16 + m][k0 + 1];
            bv.x = m12(k0 + 0, m);
            bv.y = m12(k0 + 1, m);
            acc = __builtin_amdgcn_wmma_f32_16x16x4_f32(
                false, av, false, bv, (short)0, acc, false, false);
        }
        // C/D layout: VGPR r holds row (r + 8*half), col = lane mod 16
        #pragma unroll
        for (int r = 0; r < 8; ++r) Ybuf[wave][g][r + 8 * half][m] = acc[r];
    }
    __syncthreads();

    // ---- energy_e = (1/9) * dot(Y_row(e), d_e)  (== diag of Y X^T) ----
    float energy = 0.0f;
    #pragma unroll
    for (int k = 0; k < 12; ++k) energy += Ybuf[wave][half][m][k] * d12[k];
    float val = energy * (1.0f / 9.0f) * weight;

    // ---- wave32 reduction, one atomic per wave ----
    #pragma unroll
    for (int off = 16; off > 0; off >>= 1) val += __shfl_xor(val, off, 32);
    if (lane == 0) atomicAdd(out, val);
}

extern "C" void kernel_launch(void* const* d_in, const int* in_sizes, int n_in,
                              void* d_out, int out_size, void* d_ws, size_t ws_size,
                              hipStream_t stream) {
    const float* tp    = (const float*)d_in[0];  // (F,3,3)
    const float* rot   = (const float*)d_in[1];  // (F,3,3)
    const float* verts = (const float*)d_in[2];  // (1,V,3)
    const int*   faces = (const int*)d_in[3];    // (3,F)
    const int*   edges = (const int*)d_in[4];    // (E,2)
    const int*   fids  = (const int*)d_in[5];    // (E,2)
    float* out = (float*)d_out;

    const int F = in_sizes[0] / 9;
    const int E = in_sizes[4] / 2;

    zero_out_kernel<<<1, 64, 0, stream>>>(out);
    const int blocks = (E + BLOCK - 1) / BLOCK;
    prism_loss_kernel<<<blocks, BLOCK, 0, stream>>>(tp, rot, verts, faces, edges,
                                                    fids, out, E, F);
}